// FinancialModel_39067022524974
// MI455X (gfx1250) — compile-verified
//
#include <hip/hip_runtime.h>
#include <hip/hip_bf16.h>
#include <math.h>

typedef float v2f __attribute__((ext_vector_type(2)));
typedef float v8f __attribute__((ext_vector_type(8)));

#define B_SZ    32
#define C_IN_0  8
#define C_CH    128      // C_MID == C_OUT == 128
#define L_SZ    512
#define T_STEPS 50
#define K_F     26
#define DT_C    0.5f

#define LDS_MAX (66 * (C_CH + 4))   // 66 l-positions (64 + halo) x padded channel stride

// Async global->LDS copy of one dword per active lane (gfx1250).
// lds_off: byte offset in LDS (low 32 bits of a generic &shared[] pointer);
// gp: 64-bit global address. Tracked by ASYNCcnt.
__device__ __forceinline__ void async_ld_b32(unsigned int lds_off, const void* gp)
{
  asm volatile("global_load_async_to_lds_b32 %0, %1, off"
               :: "v"(lds_off), "v"(gp)
               : "memory");
}
__device__ __forceinline__ void wait_async0()
{
  asm volatile("s_wait_asynccnt 0x0" ::: "memory");
}

// =====================================================================
// Conv1d, kernel=3, SAME (zero pad), via V_WMMA_F32_16X16X4_F32.
// Block = 128 threads (4 waves); block tile = 64 (C_out) x 64 (L);
// LDS activation tile staged transposed as xs[l_local][ci], ci-stride
// CP = Cin+4 (bank-conflict-free b64 B-fragment loads, 8B aligned).
//
// Fragments (f32 16x16x4), half = lane>>4, lm = lane&15:
//   A (16x4):  VGPR v -> W[co0+lm][ci + half*2 + v]          (2 x b32 global)
//   B (4x16):  VGPR v -> x[ci + half*2 + v][l]                (1 x ds b64)
//   C (16x16): VGPR r -> row co0 + r + half*8, col l0 + lm
// grid = (L/64, Cout/64, B), block = 128.  act: 0=none 1=relu 2=tanh
// =====================================================================
__global__ void conv1d_k3_wmma(const float* __restrict__ x,
                               const float* __restrict__ w,
                               const float* __restrict__ bias,
                               float* __restrict__ y,
                               int Cin, int act)
{
  __shared__ float xs[LDS_MAX];

  const int tid  = threadIdx.x;
  const int lane = tid & 31;
  const int wave = tid >> 5;
  const int half = lane >> 4;
  const int lm   = lane & 15;
  const int l0   = blockIdx.x * 64;
  const int co0  = blockIdx.y * 64 + wave * 16;
  const int b    = blockIdx.z;
  const int CP   = Cin + 4;

  // ---- stage x tile into LDS (transposed) via async DMA; zero halo ----
  {
    const float* xb = x + (size_t)b * Cin * L_SZ;
    const int nel = 66 * Cin;
    for (int idx = tid; idx < nel; idx += 128) {
      const int ll = idx % 66;          // local l: global l = l0 - 1 + ll
      const int ci = idx / 66;
      const int gl = l0 - 1 + ll;
      const unsigned int loff =
          (unsigned int)(uintptr_t)(&xs[ll * CP + ci]);
      if (gl >= 0 && gl < L_SZ) {
        async_ld_b32(loff, xb + (size_t)ci * L_SZ + gl);
      } else {
        xs[ll * CP + ci] = 0.0f;        // zero-pad halo (block edges only)
      }
    }
  }
  wait_async0();
  __syncthreads();

  v8f acc[4] = {};
  const float* wrow = w + ((size_t)(co0 + lm) * Cin + half * 2) * 3;

  for (int tap = 0; tap < 3; ++tap) {
    for (int ci = 0; ci < Cin; ci += 4) {
      const int k = ci + half * 2;
      v2f a;
      a.x = wrow[(size_t)ci * 3 + tap];
      a.y = wrow[(size_t)ci * 3 + tap + 3];
      // batch all four B fragments (DS clause), then back-to-back WMMAs
      const v2f b0 = *(const v2f*)&xs[(lm + tap) * CP + k];
      const v2f b1 = *(const v2f*)&xs[(16 + lm + tap) * CP + k];
      const v2f b2 = *(const v2f*)&xs[(32 + lm + tap) * CP + k];
      const v2f b3 = *(const v2f*)&xs[(48 + lm + tap) * CP + k];
      acc[0] = __builtin_amdgcn_wmma_f32_16x16x4_f32(false, a, false, b0, (short)0, acc[0], false, false);
      acc[1] = __builtin_amdgcn_wmma_f32_16x16x4_f32(false, a, false, b1, (short)0, acc[1], false, false);
      acc[2] = __builtin_amdgcn_wmma_f32_16x16x4_f32(false, a, false, b2, (short)0, acc[2], false, false);
      acc[3] = __builtin_amdgcn_wmma_f32_16x16x4_f32(false, a, false, b3, (short)0, acc[3], false, false);
    }
  }

#pragma unroll
  for (int n = 0; n < 4; ++n) {
#pragma unroll
    for (int r = 0; r < 8; ++r) {
      const int co = co0 + r + half * 8;
      const int l  = l0 + n * 16 + lm;
      float v = acc[n][r] + bias[co];
      if (act == 1)      v = fmaxf(v, 0.0f);
      else if (act == 2) v = tanhf(v);
      y[((size_t)b * C_CH + co) * L_SZ + l] = v;
    }
  }
}

// =====================================================================
// One Euler step:
//   spring = tanh(circ_conv(hy))            (circular pad, no bias)
//   hz    += DT*(spring - omega*hy - alpha*hz)   (hz == 0 when first)
//   hy    += DT*hz   -> written to this step's y_seq slice
// Same LDS tiling, circular halo (all addresses valid -> pure async fill);
// hy for the pointwise update re-read from the staged LDS tile.
// grid = (L/64, C/64, B), block = 128
// =====================================================================
__global__ void ode_step_wmma(const float* __restrict__ hy_in, size_t in_bs,
                              float* __restrict__ hy_out, size_t out_bs,
                              const float* __restrict__ wy,
                              const float* __restrict__ omega,
                              const float* __restrict__ alpha,
                              float* __restrict__ hz,
                              int first)
{
  __shared__ float xs[LDS_MAX];

  const int tid  = threadIdx.x;
  const int lane = tid & 31;
  const int wave = tid >> 5;
  const int half = lane >> 4;
  const int lm   = lane & 15;
  const int l0   = blockIdx.x * 64;
  const int co0  = blockIdx.y * 64 + wave * 16;
  const int b    = blockIdx.z;
  const int CP   = C_CH + 4;

  // ---- stage hy tile (circular halo) into LDS via async DMA ----
  {
    const float* xb = hy_in + (size_t)b * in_bs;
    const int nel = 66 * C_CH;
    for (int idx = tid; idx < nel; idx += 128) {
      const int ll = idx % 66;
      const int ci = idx / 66;
      const int gl = (l0 - 1 + ll + L_SZ) & (L_SZ - 1);   // circular
      const unsigned int loff =
          (unsigned int)(uintptr_t)(&xs[ll * CP + ci]);
      async_ld_b32(loff, xb + (size_t)ci * L_SZ + gl);
    }
  }
  wait_async0();
  __syncthreads();

  v8f acc[4] = {};
  const float* wrow = wy + ((size_t)(co0 + lm) * C_CH + half * 2) * 3;

  for (int tap = 0; tap < 3; ++tap) {
    for (int ci = 0; ci < C_CH; ci += 4) {
      const int k = ci + half * 2;
      v2f a;
      a.x = wrow[(size_t)ci * 3 + tap];
      a.y = wrow[(size_t)ci * 3 + tap + 3];
      const v2f b0 = *(const v2f*)&xs[(lm + tap) * CP + k];
      const v2f b1 = *(const v2f*)&xs[(16 + lm + tap) * CP + k];
      const v2f b2 = *(const v2f*)&xs[(32 + lm + tap) * CP + k];
      const v2f b3 = *(const v2f*)&xs[(48 + lm + tap) * CP + k];
      acc[0] = __builtin_amdgcn_wmma_f32_16x16x4_f32(false, a, false, b0, (short)0, acc[0], false, false);
      acc[1] = __builtin_amdgcn_wmma_f32_16x16x4_f32(false, a, false, b1, (short)0, acc[1], false, false);
      acc[2] = __builtin_amdgcn_wmma_f32_16x16x4_f32(false, a, false, b2, (short)0, acc[2], false, false);
      acc[3] = __builtin_amdgcn_wmma_f32_16x16x4_f32(false, a, false, b3, (short)0, acc[3], false, false);
    }
  }

#pragma unroll
  for (int n = 0; n < 4; ++n) {
#pragma unroll
    for (int r = 0; r < 8; ++r) {
      const int co = co0 + r + half * 8;
      const int l  = l0 + n * 16 + lm;
      const size_t eidx = ((size_t)b * C_CH + co) * L_SZ + l;
      const float sp  = tanhf(acc[n][r]);
      const float om  = omega[eidx];
      const float al  = alpha[eidx];
      float hyv = xs[(n * 16 + lm + 1) * CP + co];        // staged hy
      float hzv = first ? 0.0f : hz[eidx];
      hzv += DT_C * (sp - om * hyv - al * hzv);
      hyv += DT_C * hzv;
      hz[eidx] = hzv;
      hy_out[(size_t)b * out_bs + (size_t)co * L_SZ + l] = hyv;
    }
  }
}

// =====================================================================
// S[b,t,c] = mean_l y_seq[b,t,c,l]; one wave per (b,t,c), b128 loads
// =====================================================================
__global__ void mean_l_kernel(const float* __restrict__ yseq, float* __restrict__ S)
{
  const int idx  = blockIdx.x;          // flat over B*T*C
  const int lane = threadIdx.x;
  const float4* p4 = (const float4*)(yseq + (size_t)idx * L_SZ);
  float s = 0.0f;
#pragma unroll
  for (int i = lane; i < L_SZ / 4; i += 32) {
    const float4 v = p4[i];
    s += (v.x + v.y) + (v.z + v.w);
  }
#pragma unroll
  for (int off = 16; off > 0; off >>= 1) s += __shfl_down(s, off, 32);
  if (lane == 0) S[idx] = s * (1.0f / (float)L_SZ);
}

// =====================================================================
// feat[b, k*C + c] = sum_t fct[k,t] * S[b,t,c]
// =====================================================================
__global__ void fc_time_kernel(const float* __restrict__ S,
                               const float* __restrict__ fct,
                               float* __restrict__ feat)
{
  const int idx = blockIdx.x * blockDim.x + threadIdx.x;
  if (idx >= B_SZ * K_F * C_CH) return;
  const int c = idx % C_CH;
  const int k = (idx / C_CH) % K_F;
  const int b = idx / (C_CH * K_F);
  float s = 0.0f;
  for (int t = 0; t < T_STEPS; ++t)
    s += fct[k * T_STEPS + t] * S[((size_t)b * T_STEPS + t) * C_CH + c];
  feat[(size_t)b * (K_F * C_CH) + k * C_CH + c] = s;
}

// =====================================================================
// Fused MLP readout: one block (64 threads) per batch row.
// =====================================================================
__global__ void mlp_kernel(const float* __restrict__ feat,
                           const float* __restrict__ rw1, const float* __restrict__ rb1,
                           const float* __restrict__ rw2, const float* __restrict__ rb2,
                           const float* __restrict__ rw3, const float* __restrict__ rb3,
                           float* __restrict__ logits)
{
  __shared__ float h1[64];
  __shared__ float h2[32];
  const int b   = blockIdx.x;
  const int tid = threadIdx.x;
  const int KC  = K_F * C_CH;   // 3328

  {
    float s = rb1[tid];
    const float* fr = feat + (size_t)b * KC;
    const float* wr = rw1 + (size_t)tid * KC;
    for (int i = 0; i < KC; ++i) s += fr[i] * wr[i];
    h1[tid] = fmaxf(s, 0.0f);
  }
  __syncthreads();
  if (tid < 32) {
    float s = rb2[tid];
    const float* wr = rw2 + (size_t)tid * 64;
    for (int i = 0; i < 64; ++i) s += h1[i] * wr[i];
    h2[tid] = fmaxf(s, 0.0f);
  }
  __syncthreads();
  if (tid == 0) {
    float s = rb3[0];
    for (int i = 0; i < 32; ++i) s += h2[i] * rw3[i];
    logits[b] = s;
  }
}

// =====================================================================
extern "C" void kernel_launch(void* const* d_in, const int* in_sizes, int n_in,
                              void* d_out, int out_size, void* d_ws, size_t ws_size,
                              hipStream_t stream)
{
  (void)in_sizes; (void)n_in; (void)out_size; (void)ws_size;

  const float* x   = (const float*)d_in[0];
  const float* ow1 = (const float*)d_in[1];  const float* ob1 = (const float*)d_in[2];
  const float* ow2 = (const float*)d_in[3];  const float* ob2 = (const float*)d_in[4];
  const float* ow3 = (const float*)d_in[5];  const float* ob3 = (const float*)d_in[6];
  const float* aw1 = (const float*)d_in[7];  const float* ab1 = (const float*)d_in[8];
  const float* aw2 = (const float*)d_in[9];  const float* ab2 = (const float*)d_in[10];
  const float* aw3 = (const float*)d_in[11]; const float* ab3 = (const float*)d_in[12];
  const float* hw1 = (const float*)d_in[13]; const float* hb1 = (const float*)d_in[14];
  const float* hw2 = (const float*)d_in[15]; const float* hb2 = (const float*)d_in[16];
  const float* hw3 = (const float*)d_in[17]; const float* hb3 = (const float*)d_in[18];
  const float* hw4 = (const float*)d_in[19]; const float* hb4 = (const float*)d_in[20];
  const float* wy  = (const float*)d_in[21];
  const float* fct = (const float*)d_in[22];
  const float* rw1 = (const float*)d_in[23]; const float* rb1 = (const float*)d_in[24];
  const float* rw2 = (const float*)d_in[25]; const float* rb2 = (const float*)d_in[26];
  const float* rw3 = (const float*)d_in[27]; const float* rb3 = (const float*)d_in[28];

  float* ws = (float*)d_ws;
  const size_t BUF = (size_t)B_SZ * C_CH * L_SZ;     // 2,097,152 elems = 8 MB
  float* t0    = ws + 0 * BUF;
  float* t1    = ws + 1 * BUF;
  float* omega = ws + 2 * BUF;
  float* alpha = ws + 3 * BUF;
  float* hy0   = ws + 4 * BUF;
  float* hz    = ws + 5 * BUF;
  float* S     = ws + 6 * BUF;                        // B*T*C
  float* feat  = S + (size_t)B_SZ * T_STEPS * C_CH;   // B*K*C

  float* logits = (float*)d_out;                      // first 32 floats
  float* yseq   = logits + B_SZ;                      // (B,T,C,L)

  dim3 cgrid(L_SZ / 64, C_CH / 64, B_SZ);
  dim3 cblk(128);

  // omega encoder
  conv1d_k3_wmma<<<cgrid, cblk, 0, stream>>>(x,  ow1, ob1, t0,    C_IN_0, 1);
  conv1d_k3_wmma<<<cgrid, cblk, 0, stream>>>(t0, ow2, ob2, t1,    C_CH,   1);
  conv1d_k3_wmma<<<cgrid, cblk, 0, stream>>>(t1, ow3, ob3, omega, C_CH,   1);
  // alpha encoder
  conv1d_k3_wmma<<<cgrid, cblk, 0, stream>>>(x,  aw1, ab1, t0,    C_IN_0, 1);
  conv1d_k3_wmma<<<cgrid, cblk, 0, stream>>>(t0, aw2, ab2, t1,    C_CH,   1);
  conv1d_k3_wmma<<<cgrid, cblk, 0, stream>>>(t1, aw3, ab3, alpha, C_CH,   1);
  // hy encoder (last layer tanh)
  conv1d_k3_wmma<<<cgrid, cblk, 0, stream>>>(x,  hw1, hb1, t0,    C_IN_0, 1);
  conv1d_k3_wmma<<<cgrid, cblk, 0, stream>>>(t0, hw2, hb2, t1,    C_CH,   1);
  conv1d_k3_wmma<<<cgrid, cblk, 0, stream>>>(t1, hw3, hb3, t0,    C_CH,   1);
  conv1d_k3_wmma<<<cgrid, cblk, 0, stream>>>(t0, hw4, hb4, hy0,   C_CH,   2);

  // 50 oscillator steps; hy state lives in the y_seq output slices
  const size_t CL   = (size_t)C_CH * L_SZ;
  const size_t YSBS = (size_t)T_STEPS * CL;           // batch stride inside y_seq
  for (int t = 0; t < T_STEPS; ++t) {
    const float* hin = (t == 0) ? hy0 : (yseq + (size_t)(t - 1) * CL);
    const size_t in_bs = (t == 0) ? CL : YSBS;
    ode_step_wmma<<<cgrid, cblk, 0, stream>>>(hin, in_bs,
                                              yseq + (size_t)t * CL, YSBS,
                                              wy, omega, alpha, hz, (t == 0) ? 1 : 0);
  }

  // readout
  mean_l_kernel<<<B_SZ * T_STEPS * C_CH, 32, 0, stream>>>(yseq, S);
  {
    const int n = B_SZ * K_F * C_CH;
    fc_time_kernel<<<(n + 255) / 256, 256, 0, stream>>>(S, fct, feat);
  }
  mlp_kernel<<<B_SZ, 64, 0, stream>>>(feat, rw1, rb1, rw2, rb2, rw3, rb3, logits);
}